// TemporalAttention_29016799051923
// MI455X (gfx1250) — compile-verified
//
#include <hip/hip_runtime.h>
#include <cstdint>

typedef __attribute__((ext_vector_type(2))) float v2f;
typedef __attribute__((ext_vector_type(8))) float v8f;

// Problem constants
#define BN 4096
#define FN 2048
#define TN 16
#define KN 8

// Pipeline constants
#define FC    32                  // f-chunk per LDS stage
#define NCH   (FN / FC)           // 64 chunks
#define RS    (FC * TN + 4)       // 516: h row stride (floats), 16B aligned, bank step 4
#define HSZ   (16 * RS)           // 8256 floats: h region per buffer
#define WSTR  (FC + 4)            // 36: W row stride (floats), 16B aligned, conflict-free n-banks
#define WSZ   (TN * KN * WSTR)    // 4608 floats: W region per buffer
#define BUFF  (HSZ + WSZ)         // 12864 floats per buffer (~50.3 KB)
#define SH_FLOATS (2 * BUFF)      // double-buffered: 25728 floats (~100.5 KB) -> 3 WGs/WGP

__global__ __launch_bounds__(512)
void temporal_head_wmma(const float* __restrict__ h,   // [B, F, T]
                        const float* __restrict__ w,   // [K, T, F]
                        const float* __restrict__ bias,// [K, T]
                        float* __restrict__ out)       // [B, T, K]
{
    __shared__ float sh[SH_FLOATS];

    const int tid  = threadIdx.x;
    const int lane = tid & 31;
    const int t    = tid >> 5;          // wave id -> timestep (16 waves)
    const int b0   = blockIdx.x * 16;   // 16 batch rows per workgroup

    const int r    = lane & 15;         // A fragment M index
    const int half = lane >> 4;         // K-slot pair selector
    const int n    = lane & 15;         // output column (k) for B/C fragments
    const int n7   = n & 7;             // lanes n>=8 duplicate; their C columns discarded

    // ---- chunk-invariant async-transfer addresses (both buffers) ----
    // h: 16 rows x FC f x 16 t = 2048 float4 lane-transfers/chunk, 4 per thread
    unsigned           h_l0[4], h_l1[4];
    unsigned long long h_g[4];
    #pragma unroll
    for (int ii = 0; ii < 4; ++ii) {
        const int i   = tid + ii * 512;
        const int q   = i & 3;                 // t-quad
        const int fo  = (i >> 2) & (FC - 1);
        const int rr  = i >> 7;
        const int off = rr * RS + fo * 16 + 4 * q;
        h_l0[ii] = (unsigned)(uintptr_t)&sh[off];
        h_l1[ii] = (unsigned)(uintptr_t)&sh[off + BUFF];
        h_g[ii]  = (unsigned long long)(uintptr_t)
                   (h + ((size_t)(b0 + rr) * FN + fo) * TN + 4 * q);
    }
    // W: K x T x FC = 1024 float4 lane-transfers/chunk, 2 per thread
    unsigned           w_l0[2], w_l1[2];
    unsigned long long w_g[2];
    #pragma unroll
    for (int ii = 0; ii < 2; ++ii) {
        const int i   = tid + ii * 512;
        const int fq  = i & 7;                 // float4 within f-chunk
        const int tt  = (i >> 3) & 15;
        const int kk  = i >> 7;
        const int off = HSZ + (tt * KN + kk) * WSTR + 4 * fq;
        w_l0[ii] = (unsigned)(uintptr_t)&sh[off];
        w_l1[ii] = (unsigned)(uintptr_t)&sh[off + BUFF];
        w_g[ii]  = (unsigned long long)(uintptr_t)
                   (w + (size_t)(kk * TN + tt) * FN + 4 * fq);
    }

    // Per-wave LDS fragment base pointers
    const float* __restrict__ hA0 = &sh[r * RS + half * 32 + t];
    const float* __restrict__ hA1 = hA0 + BUFF;
    const float* __restrict__ wB0 = &sh[HSZ + (t * KN + n7) * WSTR + half * 2];
    const float* __restrict__ wB1 = wB0 + BUFF;

    // Per-thread prefetch cursor into the h stream (global addrspace pointer)
    const float* hpf = h + ((size_t)(b0 + (tid >> 7)) * FN + ((tid >> 2) & (FC - 1))) * TN;

    v8f acc = {0.f, 0.f, 0.f, 0.f, 0.f, 0.f, 0.f, 0.f};

    // Issue one chunk's DMA: 6 async b128 lane-transfers per thread
    auto issue = [&](int ch, const unsigned* hl, const unsigned* wl) {
        const unsigned long long hof = (unsigned long long)ch * (FC * TN * 4); // bytes
        const unsigned long long wof = (unsigned long long)ch * (FC * 4);      // bytes
        #pragma unroll
        for (int ii = 0; ii < 4; ++ii) {
            asm volatile("global_load_async_to_lds_b128 %0, %1, off"
                         :: "v"(hl[ii]), "v"(h_g[ii] + hof) : "memory");
        }
        #pragma unroll
        for (int ii = 0; ii < 2; ++ii) {
            asm volatile("global_load_async_to_lds_b128 %0, %1, off"
                         :: "v"(wl[ii]), "v"(w_g[ii] + wof) : "memory");
        }
    };

    // 8 fp32 WMMAs over one chunk's K-dim (f), 4 f-values each
    auto compute = [&](const float* __restrict__ hAp, const float* __restrict__ wBp) {
        #pragma unroll
        for (int c0 = 0; c0 < FC; c0 += 4) {
            v2f a, bf;
            a.x  = hAp[c0 * 16];          // K = c0 + 2*half
            a.y  = hAp[c0 * 16 + 16];     // K = c0 + 2*half + 1
            bf.x = wBp[c0];
            bf.y = wBp[c0 + 1];
            acc = __builtin_amdgcn_wmma_f32_16x16x4_f32(
                false, a, false, bf, (short)0, acc, false, false);
        }
    };

    // ---- double-buffered pipeline: chunk c+1 DMA in flight under chunk c compute ----
    issue(0, h_l0, w_l0);
    for (int ch = 0; ch < NCH; ch += 2) {
        // even chunk (buffer 0); odd partner always exists (NCH even)
        issue(ch + 1, h_l1, w_l1);
        if (ch + 2 < NCH) {
            __builtin_prefetch(hpf + (size_t)(ch + 2) * (FC * TN), 0, 1);
        }
        asm volatile("s_wait_asynccnt 0x6" ::: "memory");   // chunk ch landed (in-order)
        __syncthreads();
        compute(hA0, wB0);
        __syncthreads();

        // odd chunk (buffer 1)
        if (ch + 2 < NCH) {
            issue(ch + 2, h_l0, w_l0);
            asm volatile("s_wait_asynccnt 0x6" ::: "memory");
        } else {
            asm volatile("s_wait_asynccnt 0x0" ::: "memory");
        }
        __syncthreads();
        compute(hA1, wB1);
        __syncthreads();
    }

    // ---- epilogue: bias + sigmoid, store [B, T, K] ----
    const float bv = bias[n7 * TN + t];
    #pragma unroll
    for (int j = 0; j < 8; ++j) {
        const int m = j + 8 * half;                   // C layout: VGPR j -> M = j (+8 for upper half)
        const float x = acc[j] + bv;
        const float y = 1.0f / (1.0f + __expf(-x));
        if (n < KN) {
            out[((size_t)(b0 + m) * TN + t) * KN + n] = y;
        }
    }
}

extern "C" void kernel_launch(void* const* d_in, const int* in_sizes, int n_in,
                              void* d_out, int out_size, void* d_ws, size_t ws_size,
                              hipStream_t stream) {
    (void)in_sizes; (void)n_in; (void)d_ws; (void)ws_size; (void)out_size;
    const float* h    = (const float*)d_in[0];   // [B, F, T]
    const float* w    = (const float*)d_in[1];   // [K, T, F]
    const float* bias = (const float*)d_in[2];   // [K, T]
    float* out        = (float*)d_out;           // [B, T, K]

    dim3 grid(BN / 16);
    dim3 block(512);
    temporal_head_wmma<<<grid, block, 0, stream>>>(h, w, bias, out);
}